// DAC_structure_1322849927625
// MI455X (gfx1250) — compile-verified
//
#include <hip/hip_runtime.h>

typedef __attribute__((ext_vector_type(2))) float v2f;
typedef __attribute__((ext_vector_type(4))) float v4f;
typedef __attribute__((ext_vector_type(8))) float v8f;

#define LSTR 33  // LDS row stride (padding -> conflict-free column access)

// One block per (b, h). 8 waves, wave w = channel w. Each wave computes
// S = softmax( (Q K^T) / 8 ) for its channel via V_WMMA_F32_16X16X4_F32,
// then the block averages the 8 channel maps into ws.
__global__ __launch_bounds__(256) void dual_attn_scores(
    const float* __restrict__ Q, const float* __restrict__ K,
    float* __restrict__ Mout)
{
    __shared__ float S[8][32][LSTR];

    const int b    = blockIdx.x >> 3;
    const int h    = blockIdx.x & 7;
    const int tid  = threadIdx.x;
    const int wave = tid >> 5;     // channel index
    const int lane = tid & 31;
    const int half = lane >> 4;    // hi/lo 16 lanes (K split for 16x16x4 frags)
    const int l15  = lane & 15;

    const int bc = b * 8 + wave;                    // flattened B*channel
    const float* qb = Q + (size_t)bc * (32 * 8 * 64) + h * 64;  // row stride 512 f32
    const float* kb = K + (size_t)bc * (32 * 8 * 64) + h * 64;

    // 32x32 output = 2x2 tiles of 16x16; K-dim 64 = 16 steps of 4
    #pragma unroll
    for (int mi = 0; mi < 2; ++mi) {
        #pragma unroll
        for (int ni = 0; ni < 2; ++ni) {
            v8f acc = {};
            // A frag: lane holds Q[mi*16 + l15][kbase + 2*half + {0,1}]
            // B frag: lane holds K[ni*16 + l15][kbase + 2*half + {0,1}]  (B[k][n]=K[n][k])
            const float* qrow = qb + (mi * 16 + l15) * 512 + 2 * half;
            const float* krow = kb + (ni * 16 + l15) * 512 + 2 * half;
            #pragma unroll
            for (int kk = 0; kk < 16; ++kk) {
                v2f a  = *(const v2f*)(qrow + kk * 4);
                v2f bf = *(const v2f*)(krow + kk * 4);
                acc = __builtin_amdgcn_wmma_f32_16x16x4_f32(
                    /*neg_a=*/false, a, /*neg_b=*/false, bf,
                    /*c_mod=*/(short)0, acc, /*reuse_a=*/false, /*reuse_b=*/false);
            }
            // C layout: VGPR r -> row (half ? r+8 : r), col = l15
            #pragma unroll
            for (int r = 0; r < 8; ++r) {
                int m = mi * 16 + (half ? r + 8 : r);
                int n = ni * 16 + l15;
                S[wave][m][n] = acc[r] * 0.125f;   // scale = 1/sqrt(64)
            }
        }
    }
    __syncthreads();

    // Row softmax: lane handles row `lane` of channel `wave` (stride-33 -> no bank conflicts)
    {
        float* row = S[wave][lane];
        float mx = row[0];
        #pragma unroll
        for (int j = 1; j < 32; ++j) mx = fmaxf(mx, row[j]);
        float sum = 0.f;
        #pragma unroll
        for (int j = 0; j < 32; ++j) { float e = __expf(row[j] - mx); row[j] = e; sum += e; }
        float inv = 1.0f / sum;
        #pragma unroll
        for (int j = 0; j < 32; ++j) row[j] *= inv;
    }
    __syncthreads();

    // Channel mean -> ws: 1024 outputs, 4 per thread
    float* out = Mout + (size_t)(b * 8 + h) * 1024;
    #pragma unroll
    for (int t = 0; t < 4; ++t) {
        int idx = tid + t * 256;
        int r = idx >> 5, c = idx & 31;
        float s = 0.f;
        #pragma unroll
        for (int ch = 0; ch < 8; ++ch) s += S[ch][r][c];
        out[idx] = s * 0.125f;                      // mean over 8 channels
    }
}

// Streaming expansion: 512 MiB of output, one float4 NT store per thread.
// out[sel][bh][i][j]:  sel=0 (ps): element-repeat  m_ps[bh][i>>5][j>>5]
//                      sel=1 (pn): block-tile      m_pn[bh][i&31][j&31]
__global__ __launch_bounds__(256) void expand_kernel(
    const float* __restrict__ ws, v4f* __restrict__ out)
{
    const long long g = (long long)blockIdx.x * 256 + threadIdx.x; // float4 index
    const int sel = (int)(g >> 24);          // 2^24 float4 per matrix
    const int rem = (int)(g & 0xFFFFFF);
    const int bh  = rem >> 18;               // 2^18 float4 per (b,h) map
    const int r2  = rem & 0x3FFFF;
    const int i   = r2 >> 8;                 // 256 float4 per row
    const int j   = (r2 & 255) << 2;         // starting column

    v4f v;
    if (sel == 0) {
        float m = ws[bh * 1024 + ((i >> 5) << 5) + (j >> 5)];
        v = (v4f){m, m, m, m};
    } else {
        // (j&31) is a multiple of 4 and stays within one 32-block -> aligned v4 load
        v = *(const v4f*)(ws + 65536 + bh * 1024 + ((i & 31) << 5) + (j & 31));
    }
    __builtin_nontemporal_store(v, out + g);
}

extern "C" void kernel_launch(void* const* d_in, const int* in_sizes, int n_in,
                              void* d_out, int out_size, void* d_ws, size_t ws_size,
                              hipStream_t stream) {
    const float* qps = (const float*)d_in[0];
    const float* qpn = (const float*)d_in[1];
    const float* kps = (const float*)d_in[2];
    const float* kpn = (const float*)d_in[3];
    float* ws = (float*)d_ws;                 // [2][8][8][32][32] = 512 KB

    dual_attn_scores<<<64, 256, 0, stream>>>(qps, kps, ws);          // m_ps
    dual_attn_scores<<<64, 256, 0, stream>>>(qpn, kpn, ws + 65536);  // m_pn

    // 2 * 8*8*1024*1024 floats / 4 per store = 33,554,432 float4 -> 131072 blocks
    expand_kernel<<<131072, 256, 0, stream>>>(ws, (v4f*)d_out);
}